// MNERHead_74637941670056
// MI455X (gfx1250) — compile-verified
//
#include <hip/hip_runtime.h>
#include <hip/hip_bf16.h>

// Reference shapes: B=128, S=512, D=768, N=9
#define BB 128
#define SS 512
#define DD 768
#define NN 9

typedef __attribute__((ext_vector_type(2))) float v2f;
typedef __attribute__((ext_vector_type(8))) float v8f;

// ---------------------------------------------------------------------------
// Kernel 1: logits[b,s,n] = features[b,s,:] . W[n,:] + bias[n]
// Tall-skinny GEMM (65536 x 768) x (768 x 9->16) via V_WMMA_F32_16X16X4_F32.
// One wave per 16-row M-tile; 8 waves per block; 4096 tiles -> 512 blocks.
// ---------------------------------------------------------------------------
__global__ __launch_bounds__(256) void logits_wmma_kernel(
    const float* __restrict__ feat,   // [65536, 768]
    const float* __restrict__ W,      // [9, 768]
    const float* __restrict__ bias,   // [9]
    float* __restrict__ logits)       // [65536, 9]
{
    const int lane = threadIdx.x & 31;
    const int wave = threadIdx.x >> 5;
    const int tile = blockIdx.x * 8 + wave;     // 0..4095
    const int R    = tile * 16;                 // row base

    // A-fragment addressing (16x4 f32): lanes 0-15 hold K=kb..kb+1 of row m,
    // VGPR0 = K+0, VGPR1 = K+1; lane half selects K offset 0 or 2.
    const int m  = lane & 15;
    const int kb = (lane >> 4) << 1;            // 0 or 2
    // B-fragment: lane covers column n of the 4x16 B tile (B = W^T slice).
    const int n  = lane & 15;
    const int nc = (n < NN) ? n : (NN - 1);
    const float sel = (n < NN) ? 1.0f : 0.0f;   // zero the N-padding lanes

    const float* arow = feat + (size_t)(R + m) * DD + kb;
    const float* brow = W    + (size_t)nc * DD + kb;

    v8f acc = {};
#pragma unroll 4
    for (int k = 0; k < DD; k += 4) {
        v2f a = *(const v2f*)(arow + k);
        v2f b = *(const v2f*)(brow + k);
        b *= sel;
        // D = A(16x4) * B(4x16) + C ; 8 args: neg_a, A, neg_b, B, c_mod, C, reuseA, reuseB
        acc = __builtin_amdgcn_wmma_f32_16x16x4_f32(
            false, a, false, b, (short)0, acc, false, false);
    }

    // C/D layout: VGPR r -> M = r + 8*(lane>>4), N = lane&15
    if (n < NN) {
        const float bb = bias[n];
        const int mbase = (lane >> 4) * 8;
#pragma unroll
        for (int r = 0; r < 8; ++r) {
            const int row = R + mbase + r;
            logits[(size_t)row * NN + n] = acc[r] + bb;
        }
    }
}

// ---------------------------------------------------------------------------
// Kernel 2: CRF forward scan (denominator). One wave per batch, lane j owns
// state j (j<9). Alpha broadcast via lane shuffles each step.
// ---------------------------------------------------------------------------
__global__ __launch_bounds__(32) void crf_forward_kernel(
    const float* __restrict__ logits,   // [B,S,N]
    const unsigned char* __restrict__ mask, // [B,S] bool bytes
    const float* __restrict__ start_t,  // [N]
    const float* __restrict__ end_t,    // [N]
    const float* __restrict__ trans,    // [N,N]
    float* __restrict__ den)            // [B]
{
    const int b    = blockIdx.x;
    const int lane = threadIdx.x;
    const int j    = (lane < NN) ? lane : (NN - 1);
    const float* lb = logits + (size_t)b * SS * NN;
    const unsigned char* mk = mask + (size_t)b * SS;

    // Transition column j, kept in registers for the whole scan.
    float tcol[NN];
#pragma unroll
    for (int i = 0; i < NN; ++i) tcol[i] = trans[i * NN + j];

    float alpha = start_t[j] + lb[j];

    for (int s = 1; s < SS; ++s) {
        float ai[NN];
#pragma unroll
        for (int i = 0; i < NN; ++i) ai[i] = __shfl(alpha, i, 32);

        float v[NN];
        float mx = -3.4e38f;
#pragma unroll
        for (int i = 0; i < NN; ++i) {
            v[i] = ai[i] + tcol[i];
            mx = fmaxf(mx, v[i]);
        }
        float sum = 0.0f;
#pragma unroll
        for (int i = 0; i < NN; ++i) sum += __expf(v[i] - mx);

        const float nxt = mx + __logf(sum) + lb[s * NN + j];
        alpha = (mk[s] != 0) ? nxt : alpha;
    }

    // Final logsumexp over states of alpha + end_t.
    const float fv = alpha + end_t[j];
    float av[NN];
#pragma unroll
    for (int i = 0; i < NN; ++i) av[i] = __shfl(fv, i, 32);
    if (lane == 0) {
        float mx = av[0];
#pragma unroll
        for (int i = 1; i < NN; ++i) mx = fmaxf(mx, av[i]);
        float sum = 0.0f;
#pragma unroll
        for (int i = 0; i < NN; ++i) sum += __expf(av[i] - mx);
        den[b] = mx + __logf(sum);
    }
}

// ---------------------------------------------------------------------------
// Kernel 3: numerator score per batch + loss reduction. One block, thread=batch.
// ---------------------------------------------------------------------------
__global__ __launch_bounds__(BB) void crf_loss_kernel(
    const float* __restrict__ logits,
    const int* __restrict__ labels,
    const unsigned char* __restrict__ mask,
    const float* __restrict__ start_t,
    const float* __restrict__ end_t,
    const float* __restrict__ trans,
    const float* __restrict__ den,
    float* __restrict__ loss)
{
    __shared__ float red[BB];
    const int b = threadIdx.x;
    const int* lab = labels + (size_t)b * SS;
    const unsigned char* mk = mask + (size_t)b * SS;
    const float* lb = logits + (size_t)b * SS * NN;

    int t0 = lab[0]; if (t0 == -100) t0 = 0;
    int prev = t0;
    float score = start_t[t0] + lb[t0];
    int seq_end = 0;  // index of last mask==1 (mask[0] assumed true per CRF convention)

    for (int s = 1; s < SS; ++s) {
        int cur = lab[s]; if (cur == -100) cur = 0;
        const bool mm = (mk[s] != 0);
        if (mm) {
            score += lb[s * NN + cur] + trans[prev * NN + cur];
            seq_end = s;
        }
        prev = cur;
    }
    int tl = lab[seq_end]; if (tl == -100) tl = 0;
    const float numerator = score + end_t[tl];
    red[b] = numerator - den[b];
    __syncthreads();

    // tree reduce 128 -> 1
    for (int off = BB / 2; off > 0; off >>= 1) {
        if (b < off) red[b] += red[b + off];
        __syncthreads();
    }
    if (b == 0) loss[0] = -red[0] / (float)BB;
}

// ---------------------------------------------------------------------------
// Host-side launcher
// ---------------------------------------------------------------------------
extern "C" void kernel_launch(void* const* d_in, const int* in_sizes, int n_in,
                              void* d_out, int out_size, void* d_ws, size_t ws_size,
                              hipStream_t stream) {
    const float*         features = (const float*)d_in[0];         // [B,S,D]
    const int*           labels   = (const int*)d_in[1];           // [B,S]
    const unsigned char* mask     = (const unsigned char*)d_in[2]; // [B,S] bool
    const float*         W        = (const float*)d_in[3];         // [N,D]
    const float*         bvec     = (const float*)d_in[4];         // [N]
    const float*         start_t  = (const float*)d_in[5];         // [N]
    const float*         end_t    = (const float*)d_in[6];         // [N]
    const float*         trans    = (const float*)d_in[7];         // [N,N]

    float* out    = (float*)d_out;
    float* loss   = out;       // out[0]
    float* logits = out + 1;   // out[1 .. 1+B*S*N)
    float* den    = (float*)d_ws;  // [B] scratch

    // 65536 rows / 16 = 4096 tiles; 8 waves per block -> 512 blocks.
    logits_wmma_kernel<<<512, 256, 0, stream>>>(features, W, bvec, logits);
    crf_forward_kernel<<<BB, 32, 0, stream>>>(logits, mask, start_t, end_t, trans, den);
    crf_loss_kernel<<<1, BB, 0, stream>>>(logits, labels, mask, start_t, end_t, trans,
                                          den, loss);
}